// RippleNet_18116172055030
// MI455X (gfx1250) — compile-verified
//
#include <hip/hip_runtime.h>
#include <math.h>

// ---------------- problem constants ----------------
#define NENT   200000
#define NREL   32
#define DIM    64
#define NHOP   2
#define NMEM   32
#define NBATCH 1024
#define NE     (NHOP * NBATCH * NMEM)    // 65536 (b,m,hop) entries
#define ALIGNR 128                       // per-relation region alignment (8 tiles)
#define PADN   (NE + NREL * ALIGNR)      // 69632
#define NBLOCKS (PADN / ALIGNR)          // 544 workgroups, 8 waves each

typedef __attribute__((ext_vector_type(2))) float v2f;
typedef __attribute__((ext_vector_type(8))) float v8f;
typedef __attribute__((ext_vector_type(4))) unsigned int u32x4;
typedef __attribute__((ext_vector_type(4))) int i32x4;
typedef __attribute__((ext_vector_type(8))) int i32x8;

// ---------------- workspace layout (4-byte units; ints then floats) ----------
#define WS_COUNTERS 0                    // int[32]
#define WS_CURSORS  32                   // int[32]
#define WS_OFFSETS  64                   // int[33] (128-aligned exclusive scan)
#define WS_COMPACT  128                  // int[PADN] entry ids grouped by relation
#define WS_INT_END  (128 + PADN)         // = 69760
#define WF_RELNORM  WS_INT_END           // float[32]  sum(R_r^2)
#define WF_RH       (WF_RELNORM + 32)    // float[NE*DIM]  R@h per entry
#define WF_HRT      (WF_RH + NE * DIM)   // float[NE]      h^T R t per entry
#define WF_BCE      (WF_HRT + NE)        // float[NBATCH]
#define WF_KGE      (WF_BCE + NBATCH)    // float[NBATCH]
#define WF_L2       (WF_KGE + NBATCH)    // float[NBATCH]

// ---------------- WMMA helper: D = A(16x4,f32) * B(4x16,f32) + C ------------
__device__ inline v8f wmma_f32_4(v2f a, v2f b, v8f c) {
  return __builtin_amdgcn_wmma_f32_16x16x4_f32(false, a, false, b, (short)0, c,
                                               false, false);
}

// ---------------- phase 0: zero int region (counters/cursors/compact) -------
__global__ void rk_init(int* wsi) {
  int i = blockIdx.x * blockDim.x + threadIdx.x;
  if (i < WS_INT_END) wsi[i] = 0;
}

// ---------------- phase 1: per-relation squared norm ------------------------
__global__ __launch_bounds__(256) void rk_relnorm(const float* __restrict__ rel,
                                                  float* __restrict__ wsf) {
  __shared__ float red[256];
  int r = blockIdx.x, t = threadIdx.x;
  float s = 0.f;
  for (int i = t; i < DIM * DIM; i += 256) {
    float x = rel[(long)r * DIM * DIM + i];
    s += x * x;
  }
  red[t] = s;
  __syncthreads();
  for (int st = 128; st > 0; st >>= 1) {
    if (t < st) red[t] += red[t + st];
    __syncthreads();
  }
  if (t == 0) wsf[WF_RELNORM + r] = red[0];
}

// ---------------- phase 2: histogram of relation usage ----------------------
__global__ void rk_count(const int* __restrict__ mem_r, int* wsi) {
  int e = blockIdx.x * blockDim.x + threadIdx.x;
  if (e < NE) atomicAdd(&wsi[WS_COUNTERS + mem_r[e]], 1);
}

// ---------------- phase 3: 128-aligned exclusive scan -----------------------
__global__ void rk_scan(int* wsi) {
  if (blockIdx.x == 0 && threadIdx.x == 0) {
    int acc = 0;
    for (int r = 0; r < NREL; ++r) {
      wsi[WS_OFFSETS + r] = acc;
      acc += (wsi[WS_COUNTERS + r] + (ALIGNR - 1)) & ~(ALIGNR - 1);
    }
    wsi[WS_OFFSETS + NREL] = acc;
  }
}

// ---------------- phase 4: scatter entry ids into relation groups -----------
__global__ void rk_scatter(const int* __restrict__ mem_r, int* wsi) {
  int e = blockIdx.x * blockDim.x + threadIdx.x;
  if (e < NE) {
    int r = mem_r[e];
    int pos = wsi[WS_OFFSETS + r] + atomicAdd(&wsi[WS_CURSORS + r], 1);
    wsi[WS_COMPACT + pos] = e;  // pad slots stay 0 (valid addresses, discarded)
  }
}

// ---------------- phase 5: grouped GEMM via V_WMMA_F32_16X16X4_F32 ----------
// 8 waves per block share one LDS-staged copy of R_r (TDM tensor load).
// Each wave owns one 16-row tile: computes Rh[16x64] (stored) and Rt[16x64]
// (dotted in-register against h -> scalar hRt per row).
__global__ __launch_bounds__(256) void rk_wmma_gemm(
    const int* __restrict__ mem_h, const int* __restrict__ mem_t,
    const float* __restrict__ item_table, const float* __restrict__ rel_table,
    const int* __restrict__ counters, const int* __restrict__ offs,
    const int* __restrict__ compact, float* __restrict__ Rh,
    float* __restrict__ hRt) {
  __shared__ float R_sh[DIM * DIM];  // 16 KB of the 320 KB WGP LDS
  int blockStart = blockIdx.x * ALIGNR;
  if (blockStart >= offs[NREL]) return;  // uniform over block
  int r = 0;
  while (r < NREL - 1 && blockStart >= offs[r + 1]) ++r;
  int rowEnd = offs[r] + counters[r];
  const float* Rg = rel_table + (long)r * DIM * DIM;
  int tid = threadIdx.x;

#if __has_builtin(__builtin_amdgcn_tensor_load_to_lds)
  // --- TDM: DMA the 16KB R_r tile into LDS (ISA 8.3/8.4 D# layout) ---
  if (tid < 32) {  // wave 0 issues; TDM ignores EXEC, per-wave instruction
    unsigned long long ga = (unsigned long long)(uintptr_t)Rg;
    unsigned int lds = (unsigned int)(uintptr_t)&R_sh[0];  // low 32b = LDS byte addr
    u32x4 g0;
    g0.x = 1u;                                   // count=1 valid descriptor
    g0.y = lds;                                  // lds_addr
    g0.z = (unsigned int)ga;                     // global_addr[31:0]
    g0.w = (unsigned int)((ga >> 32) & 0x01FFFFFFu) | (2u << 30);  // addr[56:32]|type=2
    i32x8 g1;
    g1[0] = (2 << 16);                           // workgroup_mask=0, data_size=4B
    g1[1] = (int)((4096u & 0xFFFFu) << 16);      // tensor_dim0[15:0] @bits63:48
    g1[2] = (int)((4096u >> 16) | (1u << 16));   // tensor_dim0 hi | tensor_dim1=1
    g1[3] = (int)(4096u << 16);                  // tile_dim0=4096 @bits127:112
    g1[4] = 1;                                   // tile_dim1=1
    g1[5] = 4096;                                // tensor_dim0_stride[31:0]
    g1[6] = 0;
    g1[7] = 0;
    i32x4 gz4 = {0, 0, 0, 0};                    // dims >2 unused
    i32x8 gz8 = {0, 0, 0, 0, 0, 0, 0, 0};
    __builtin_amdgcn_tensor_load_to_lds(g0, g1, gz4, gz4, gz8, 0);
    __builtin_amdgcn_s_wait_tensorcnt((short)0);
  }
#else
  for (int i = tid; i < DIM * DIM; i += 256) R_sh[i] = Rg[i];
#endif
  __syncthreads();

  int wave = tid >> 5;
  int lane = tid & 31;
  int tileStart = blockStart + wave * 16;
  if (tileStart >= rowEnd) return;  // pure padding tile; no syncs follow
  int half = lane >> 4;             // 0: K{0,1}  1: K{2,3}
  int l16 = lane & 15;

  int eA = compact[tileStart + l16];  // lane l16 owns output row l16
  int hIdx = mem_h[eA], tIdx = mem_t[eA];
  long hbase = (long)hIdx * DIM;
  long tbase = (long)tIdx * DIM;

  v8f acc_h[4] = {}, acc_t[4] = {};
  for (int k0 = 0; k0 < DIM; k0 += 4) {
    int kA = k0 + half * 2;
    v2f a_h, a_t;
    a_h.x = item_table[hbase + kA];
    a_h.y = item_table[hbase + kA + 1];
    a_t.x = item_table[tbase + kA];
    a_t.y = item_table[tbase + kA + 1];
#pragma unroll
    for (int nt = 0; nt < 4; ++nt) {
      int n = nt * 16 + l16;  // output dim i; B[K=j,N=i] = R[i*64+j]
      v2f bm;
      bm.x = R_sh[n * DIM + kA];
      bm.y = R_sh[n * DIM + kA + 1];
      acc_h[nt] = wmma_f32_4(a_h, bm, acc_h[nt]);
      acc_t[nt] = wmma_f32_4(a_t, bm, acc_t[nt]);
    }
  }

  // ---- store Rh: lane holds N=l16; VGPR v holds row M = v + half*8 ----
  bool fullTile = (tileStart + 16 <= rowEnd);
  if (fullTile) {
#pragma unroll
    for (int nt = 0; nt < 4; ++nt)
#pragma unroll
      for (int v = 0; v < 8; ++v) {
        int e = __shfl(eA, v + half * 8, 32);  // entry id of row M
        Rh[(long)e * DIM + nt * 16 + l16] = acc_h[nt][v];
      }
  } else {
#pragma unroll
    for (int nt = 0; nt < 4; ++nt)
#pragma unroll
      for (int v = 0; v < 8; ++v) {
        int row = tileStart + v + half * 8;
        int e = __shfl(eA, v + half * 8, 32);
        if (row < rowEnd) Rh[(long)e * DIM + nt * 16 + l16] = acc_h[nt][v];
      }
  }

  // ---- hRt = dot(Rt[row,:], h[row,:]) : xor-reduce within 16-lane halves ----
#pragma unroll
  for (int v = 0; v < 8; ++v) {
    int row = tileStart + v + half * 8;
    int e = __shfl(eA, v + half * 8, 32);
    long hb = (long)__shfl(hIdx, v + half * 8, 32) * DIM;
    float p = 0.f;
#pragma unroll
    for (int nt = 0; nt < 4; ++nt)
      p += acc_t[nt][v] * item_table[hb + nt * 16 + l16];
    p += __shfl_xor(p, 1, 32);
    p += __shfl_xor(p, 2, 32);
    p += __shfl_xor(p, 4, 32);
    p += __shfl_xor(p, 8, 32);
    if (l16 == 0 && row < rowEnd) hRt[e] = p;
  }
}

// ---------------- phase 6: per-batch hop-sequential pipeline ----------------
__global__ __launch_bounds__(64) void rk_batch(
    const int* __restrict__ items, const float* __restrict__ labels,
    const int* __restrict__ mem_h, const int* __restrict__ mem_r,
    const int* __restrict__ mem_t, const float* __restrict__ item_table,
    const float* __restrict__ W, const float* __restrict__ Rh,
    const float* __restrict__ hRt, const float* __restrict__ relnorm,
    float* __restrict__ bce, float* __restrict__ kge, float* __restrict__ l2p,
    float* __restrict__ scores) {
  __shared__ float v_sh[DIM], u_sh[DIM], prob[NMEM], red[DIM];
  int b = blockIdx.x, tid = threadIdx.x;
  v_sh[tid] = item_table[(long)items[b] * DIM + tid];
  __builtin_prefetch(&W[tid * DIM], 0, 3);
  float y_acc = 0.f, kge_loc = 0.f, l2_loc = 0.f;
  __syncthreads();

  for (int hop = 0; hop < NHOP; ++hop) {
    int ebase = (hop * NBATCH + b) * NMEM;
    if (tid < NMEM) {  // logits: dot(Rh[b,m], v)
      const float* rh = Rh + (long)(ebase + tid) * DIM;
      float s = 0.f;
      for (int j = 0; j < DIM; ++j) s += rh[j] * v_sh[j];
      prob[tid] = s;
    }
    __syncthreads();
    if (tid == 0) {  // softmax over M=32
      float mx = prob[0];
      for (int m = 1; m < NMEM; ++m) mx = fmaxf(mx, prob[m]);
      float sum = 0.f;
      for (int m = 0; m < NMEM; ++m) {
        float p = expf(prob[m] - mx);
        prob[m] = p;
        sum += p;
      }
      float inv = 1.f / sum;
      for (int m = 0; m < NMEM; ++m) prob[m] *= inv;
    }
    __syncthreads();
    float o = 0.f;  // o_j = sum_m t[b,m,j] * prob[m]
    for (int m = 0; m < NMEM; ++m) {
      long ti = (long)mem_t[ebase + m] * DIM;
      o += item_table[ti + tid] * prob[m];
    }
    y_acc += o;
    u_sh[tid] = v_sh[tid] + o;
    if (tid < NMEM) {  // KGE + L2 partials, one memory slot per thread
      int e = ebase + tid;
      kge_loc += 1.f / (1.f + expf(-hRt[e]));
      float c = relnorm[mem_r[e]];
      long hb = (long)mem_h[e] * DIM, tb = (long)mem_t[e] * DIM;
      for (int j = 0; j < DIM; ++j) {
        float hv = item_table[hb + j], tv = item_table[tb + j];
        c += hv * hv + tv * tv;
      }
      l2_loc += c;
    }
    __syncthreads();
    float nv = 0.f;  // v = (v + o) @ W^T
    for (int j = 0; j < DIM; ++j) nv += u_sh[j] * W[tid * DIM + j];
    __syncthreads();
    v_sh[tid] = nv;
    __syncthreads();
  }

  red[tid] = v_sh[tid] * y_acc;  // score = sigmoid(dot(v, y))
  __syncthreads();
  for (int s = 32; s > 0; s >>= 1) {
    if (tid < s) red[tid] += red[tid + s];
    __syncthreads();
  }
  if (tid == 0) {
    float sc = 1.f / (1.f + expf(-red[0]));
    scores[b] = sc;
    float lab = labels[b];
    float lp = fmaxf(logf(sc), -100.f);
    float lnp = fmaxf(logf(1.f - sc), -100.f);
    bce[b] = lab * lp + (1.f - lab) * lnp;
  }
  __syncthreads();
  red[tid] = kge_loc;
  __syncthreads();
  for (int s = 32; s > 0; s >>= 1) {
    if (tid < s) red[tid] += red[tid + s];
    __syncthreads();
  }
  if (tid == 0) kge[b] = red[0];
  __syncthreads();
  red[tid] = l2_loc;
  __syncthreads();
  for (int s = 32; s > 0; s >>= 1) {
    if (tid < s) red[tid] += red[tid + s];
    __syncthreads();
  }
  if (tid == 0) l2p[b] = red[0];
}

// ---------------- phase 7: deterministic final reduction --------------------
__global__ __launch_bounds__(256) void rk_final(const float* __restrict__ bce,
                                                const float* __restrict__ kge,
                                                const float* __restrict__ l2p,
                                                float* __restrict__ out) {
  __shared__ float r0[256], r1[256], r2[256];
  int t = threadIdx.x;
  float a = 0.f, k = 0.f, c = 0.f;
  for (int i = t; i < NBATCH; i += 256) {
    a += bce[i];
    k += kge[i];
    c += l2p[i];
  }
  r0[t] = a; r1[t] = k; r2[t] = c;
  __syncthreads();
  for (int s = 128; s > 0; s >>= 1) {
    if (t < s) { r0[t] += r0[t + s]; r1[t] += r1[t + s]; r2[t] += r2[t + s]; }
    __syncthreads();
  }
  if (t == 0) {
    float base = -r0[0] / (float)NBATCH;
    float kgel = -0.01f * (r1[0] / (float)(NBATCH * NMEM));
    float l2l = 1e-7f * r2[0];
    out[NBATCH + 0] = base + kgel + l2l;  // loss
    out[NBATCH + 1] = base;               // base_loss
    out[NBATCH + 2] = kgel;               // kge_loss
    out[NBATCH + 3] = l2l;                // l2_loss
  }
}

// ---------------- launcher --------------------------------------------------
extern "C" void kernel_launch(void* const* d_in, const int* in_sizes, int n_in,
                              void* d_out, int out_size, void* d_ws,
                              size_t ws_size, hipStream_t stream) {
  (void)in_sizes; (void)n_in; (void)out_size; (void)ws_size;
  const int*   items      = (const int*)d_in[0];
  const float* labels     = (const float*)d_in[1];
  const int*   mem_h      = (const int*)d_in[2];
  const int*   mem_r      = (const int*)d_in[3];
  const int*   mem_t      = (const int*)d_in[4];
  const float* item_table = (const float*)d_in[5];
  const float* rel_table  = (const float*)d_in[6];
  const float* W          = (const float*)d_in[7];
  float* out = (float*)d_out;

  int*   wsi = (int*)d_ws;
  float* wsf = (float*)d_ws;
  float* Rh      = wsf + WF_RH;
  float* hRt     = wsf + WF_HRT;
  float* relnorm = wsf + WF_RELNORM;
  float* bce     = wsf + WF_BCE;
  float* kge     = wsf + WF_KGE;
  float* l2p     = wsf + WF_L2;

  rk_init<<<(WS_INT_END + 255) / 256, 256, 0, stream>>>(wsi);
  rk_relnorm<<<NREL, 256, 0, stream>>>(rel_table, wsf);
  rk_count<<<NE / 256, 256, 0, stream>>>(mem_r, wsi);
  rk_scan<<<1, 1, 0, stream>>>(wsi);
  rk_scatter<<<NE / 256, 256, 0, stream>>>(mem_r, wsi);
  rk_wmma_gemm<<<NBLOCKS, 256, 0, stream>>>(
      mem_h, mem_t, item_table, rel_table, wsi + WS_COUNTERS, wsi + WS_OFFSETS,
      wsi + WS_COMPACT, Rh, hRt);
  rk_batch<<<NBATCH, 64, 0, stream>>>(items, labels, mem_h, mem_r, mem_t,
                                      item_table, W, Rh, hRt, relnorm, bce, kge,
                                      l2p, out);
  rk_final<<<1, 256, 0, stream>>>(bce, kge, l2p, out);
}